// HeteroGNN_78426102825755
// MI455X (gfx1250) — compile-verified
//
#include <hip/hip_runtime.h>

typedef __attribute__((ext_vector_type(2))) float v2f;
typedef __attribute__((ext_vector_type(8))) float v8f;

#define NG_  50000
#define ND_  10000
#define INF_ 64
#define HID_ 128
#define EGG_ 800000
#define EGD_ 300000
#define EDG_ 300000
#define BN_EPS 1e-5f

// ---------------------------------------------------------------- utilities

__global__ void fill_kernel(float* __restrict__ p, long long n, float v) {
  long long stride = (long long)gridDim.x * blockDim.x;
  for (long long t = (long long)blockIdx.x * blockDim.x + threadIdx.x; t < n; t += stride)
    p[t] = v;
}

__global__ void degree_kernel(const int* __restrict__ dst, int E, float* __restrict__ deg) {
  int t = blockIdx.x * blockDim.x + threadIdx.x;
  if (t < E) atomicAdd(&deg[dst[t]], 1.0f);
}

// per-(edge,feature) scatter add: lanes cover consecutive features of one edge,
// so atomic targets and gather reads are contiguous within the destination row.
__global__ void scatter_add_kernel(const float* __restrict__ H,
                                   const int* __restrict__ src,
                                   const int* __restrict__ dst,
                                   long long total, int logD,
                                   float* __restrict__ S) {
  long long stride = (long long)gridDim.x * blockDim.x;
  int dmask = (1 << logD) - 1;
  for (long long t = (long long)blockIdx.x * blockDim.x + threadIdx.x; t < total; t += stride) {
    int e = (int)(t >> logD);
    int f = (int)t & dmask;
    atomicAdd(&S[((size_t)dst[e] << logD) + f], H[((size_t)src[e] << logD) + f]);
  }
}

// ---------------------------------------------------------------- WMMA GEMM
// D[N,OUT] = sum_p (A_p / max(div_p,1)) @ W_p + bias0 + bias1
// fp32 WMMA 16x16x4; one wave -> one 16x16 output tile.

__device__ inline v8f wmma_pair(v8f acc, const float* __restrict__ A,
                                const float* __restrict__ dv,
                                const float* __restrict__ W,
                                int row, int col, int K, int OUT, int kb) {
  float rdiv = 1.0f;
  if (dv) rdiv = 1.0f / fmaxf(dv[row], 1.0f);
  const float* arow = A + (size_t)row * K;
  for (int k = 0; k < K; k += 4) {
    v2f a, b;
    // A 16x4 fragment: lanes 0-15 row=lane, VGPR0/1 = K {k,k+1}; lanes 16-31 = K {k+2,k+3}
    a.x = arow[k + kb] * rdiv;
    a.y = arow[k + kb + 1] * rdiv;
    // B 4x16 fragment: lanes hold columns, same K split as A
    b.x = W[(size_t)(k + kb) * OUT + col];
    b.y = W[(size_t)(k + kb + 1) * OUT + col];
    acc = __builtin_amdgcn_wmma_f32_16x16x4_f32(false, a, false, b, (short)0, acc,
                                                false, false);
  }
  return acc;
}

__global__ __launch_bounds__(128)
void gemm_wmma_kernel(const float* __restrict__ A0, const float* __restrict__ dv0, const float* __restrict__ W0,
                      const float* __restrict__ A1, const float* __restrict__ dv1, const float* __restrict__ W1,
                      const float* __restrict__ A2, const float* __restrict__ dv2, const float* __restrict__ W2,
                      const float* __restrict__ A3, const float* __restrict__ dv3, const float* __restrict__ W3,
                      const float* __restrict__ bias0, const float* __restrict__ bias1,
                      float* __restrict__ Dst, int N, int K, int OUT) {
  int lane = threadIdx.x & 31;
  int wave = threadIdx.x >> 5;
  int rowTile = blockIdx.x * 4 + wave;          // 4 waves per block
  int row0 = rowTile * 16;
  if (row0 >= N) return;                        // wave-uniform exit, EXEC stays full
  int col0 = blockIdx.y * 16;

  int rl  = lane & 15;
  int kb  = (lane >> 4) << 1;                   // 0 for lanes 0-15, 2 for lanes 16-31
  int row = row0 + rl;
  int col = col0 + rl;

  v8f acc = {};
  if (A0) acc = wmma_pair(acc, A0, dv0, W0, row, col, K, OUT, kb);
  if (A1) acc = wmma_pair(acc, A1, dv1, W1, row, col, K, OUT, kb);
  if (A2) acc = wmma_pair(acc, A2, dv2, W2, row, col, K, OUT, kb);
  if (A3) acc = wmma_pair(acc, A3, dv3, W3, row, col, K, OUT, kb);

  float bias = 0.0f;
  if (bias0) bias += bias0[col];
  if (bias1) bias += bias1[col];

  // C/D layout: VGPR i -> M = i (lanes 0-15) / 8+i (lanes 16-31), N = lane&15
  int rbase = row0 + ((lane >> 4) << 3);
#pragma unroll
  for (int i = 0; i < 8; ++i)
    Dst[(size_t)(rbase + i) * OUT + col] = acc[i] + bias;
}

// ---------------------------------------------------------------- BatchNorm

__global__ void bnstats_kernel(const float* __restrict__ H, int N, int C,
                               float* __restrict__ stats) {
  int tid = blockIdx.x * blockDim.x + threadIdx.x;
  int c  = tid % C;
  int r0 = tid / C;
  int rs = (gridDim.x * blockDim.x) / C;
  float s = 0.0f, s2 = 0.0f;
  for (int r = r0; r < N; r += rs) {
    float v = H[(size_t)r * C + c];
    s += v; s2 += v * v;
  }
  atomicAdd(&stats[c], s);
  atomicAdd(&stats[C + c], s2);
}

__global__ void bnfin_kernel(const float* __restrict__ stats,
                             const float* __restrict__ g, const float* __restrict__ b,
                             int N, int C, float* __restrict__ ss) {
  int c = blockIdx.x * blockDim.x + threadIdx.x;
  if (c >= C) return;
  float mean = stats[c] / (float)N;
  float var  = stats[C + c] / (float)N - mean * mean;
  float sc   = g[c] * rsqrtf(var + BN_EPS);
  ss[c]     = sc;
  ss[C + c] = b[c] - mean * sc;
}

__global__ void bnapply_kernel(float* __restrict__ H, long long total, int C,
                               const float* __restrict__ ss) {
  long long stride = (long long)gridDim.x * blockDim.x;
  for (long long t = (long long)blockIdx.x * blockDim.x + threadIdx.x; t < total; t += stride) {
    int c = (int)(t % C);
    H[t] = fmaxf(H[t] * ss[c] + ss[C + c], 0.0f);
  }
}

// ---------------------------------------------------------------- launcher

static inline int igrid(long long n, int tpb, int cap) {
  long long b = (n + tpb - 1) / tpb;
  if (b > cap) b = cap;
  if (b < 1) b = 1;
  return (int)b;
}

extern "C" void kernel_launch(void* const* d_in, const int* in_sizes, int n_in,
                              void* d_out, int out_size, void* d_ws, size_t ws_size,
                              hipStream_t stream) {
  (void)in_sizes; (void)n_in; (void)out_size; (void)ws_size;
  const float* xg  = (const float*)d_in[0];
  const float* xd  = (const float*)d_in[1];
  const float* Wn0 = (const float*)d_in[2];   // [3,64,128]
  const float* Ws0 = (const float*)d_in[3];
  const float* b0  = (const float*)d_in[4];   // [3,128]
  const float* Wn1 = (const float*)d_in[5];   // [3,128,128]
  const float* Ws1 = (const float*)d_in[6];
  const float* b1  = (const float*)d_in[7];
  const float* bng = (const float*)d_in[8];   // [2,2,128]
  const float* bnb = (const float*)d_in[9];
  const float* Wp  = (const float*)d_in[10];  // [2,128,64]
  const float* bp  = (const float*)d_in[11];  // [2,64]
  const int* gg_src = (const int*)d_in[12];
  const int* gg_dst = (const int*)d_in[13];
  const int* gd_src = (const int*)d_in[14];
  const int* gd_dst = (const int*)d_in[15];
  const int* dg_src = (const int*)d_in[16];
  const int* dg_dst = (const int*)d_in[17];

  float* ws = (float*)d_ws;
  size_t o = 0;
  float* degGG = ws + o; o += NG_;
  float* degDG = ws + o; o += NG_;
  float* degGD = ws + o; o += ND_;
  float* aggGG = ws + o; o += (size_t)NG_ * HID_;
  float* aggDG = ws + o; o += (size_t)NG_ * HID_;   // contiguous with aggGG
  float* aggGD = ws + o; o += (size_t)ND_ * HID_;
  float* hg0   = ws + o; o += (size_t)NG_ * HID_;
  float* hd0   = ws + o; o += (size_t)ND_ * HID_;
  float* hg1   = ws + o; o += (size_t)NG_ * HID_;
  float* hd1   = ws + o; o += (size_t)ND_ * HID_;
  float* stats = ws + o; o += 2 * HID_;
  float* ssbuf = ws + o; o += 2 * HID_;

  float* outG = (float*)d_out;
  float* outD = outG + (size_t)NG_ * INF_;

  auto zero = [&](float* p, long long n) {
    fill_kernel<<<igrid(n, 256, 32768), 256, 0, stream>>>(p, n, 0.0f);
  };
  auto scatter = [&](const float* H, const int* s, const int* d, long long E, int logD, float* S) {
    long long tot = E << logD;
    scatter_add_kernel<<<igrid(tot, 256, 120000), 256, 0, stream>>>(H, s, d, tot, logD, S);
  };
  auto bn = [&](float* H, int N, const float* g, const float* b) {
    zero(stats, 2 * HID_);
    bnstats_kernel<<<256, 256, 0, stream>>>(H, N, HID_, stats);
    bnfin_kernel<<<1, HID_, 0, stream>>>(stats, g, b, N, HID_, ssbuf);
    long long tot = (long long)N * HID_;
    bnapply_kernel<<<igrid(tot, 256, 32768), 256, 0, stream>>>(H, tot, HID_, ssbuf);
  };

  // ---- degrees (shared across both layers)
  zero(degGG, NG_ + NG_ + ND_);                 // contiguous deg buffers
  degree_kernel<<<igrid(EGG_, 256, 1 << 22), 256, 0, stream>>>(gg_dst, EGG_, degGG);
  degree_kernel<<<igrid(EDG_, 256, 1 << 22), 256, 0, stream>>>(dg_dst, EDG_, degDG);
  degree_kernel<<<igrid(EGD_, 256, 1 << 22), 256, 0, stream>>>(gd_dst, EGD_, degGD);

  // ---- layer 0 aggregation (D = 64)
  zero(aggGG, (long long)NG_ * INF_);
  zero(aggDG, (long long)NG_ * INF_);
  zero(aggGD, (long long)ND_ * INF_);
  scatter(xg, gg_src, gg_dst, EGG_, 6, aggGG);
  scatter(xd, dg_src, dg_dst, EDG_, 6, aggDG);
  scatter(xg, gd_src, gd_dst, EGD_, 6, aggGD);

  // ---- layer 0 GEMMs  (K=64, OUT=128)
  dim3 gG0((NG_ / 16 + 3) / 4, HID_ / 16);
  gemm_wmma_kernel<<<gG0, 128, 0, stream>>>(
      aggGG, degGG, Wn0 + 0 * INF_ * HID_,
      aggDG, degDG, Wn0 + 2 * INF_ * HID_,
      xg, nullptr,  Ws0 + 0 * INF_ * HID_,
      xg, nullptr,  Ws0 + 2 * INF_ * HID_,
      b0 + 0 * HID_, b0 + 2 * HID_,
      hg0, NG_, INF_, HID_);
  dim3 gD0((ND_ / 16 + 3) / 4, HID_ / 16);
  gemm_wmma_kernel<<<gD0, 128, 0, stream>>>(
      aggGD, degGD, Wn0 + 1 * INF_ * HID_,
      xd, nullptr,  Ws0 + 1 * INF_ * HID_,
      nullptr, nullptr, nullptr,
      nullptr, nullptr, nullptr,
      b0 + 1 * HID_, nullptr,
      hd0, ND_, INF_, HID_);

  bn(hg0, NG_, bng + 0 * HID_, bnb + 0 * HID_);   // layer0, gene
  bn(hd0, ND_, bng + 1 * HID_, bnb + 1 * HID_);   // layer0, disease

  // ---- layer 1 aggregation (D = 128)
  zero(aggGG, (long long)NG_ * HID_ * 2);         // aggGG + aggDG contiguous
  zero(aggGD, (long long)ND_ * HID_);
  scatter(hg0, gg_src, gg_dst, EGG_, 7, aggGG);
  scatter(hd0, dg_src, dg_dst, EDG_, 7, aggDG);
  scatter(hg0, gd_src, gd_dst, EGD_, 7, aggGD);

  // ---- layer 1 GEMMs  (K=128, OUT=128)
  gemm_wmma_kernel<<<gG0, 128, 0, stream>>>(
      aggGG, degGG, Wn1 + 0 * HID_ * HID_,
      aggDG, degDG, Wn1 + 2 * HID_ * HID_,
      hg0, nullptr, Ws1 + 0 * HID_ * HID_,
      hg0, nullptr, Ws1 + 2 * HID_ * HID_,
      b1 + 0 * HID_, b1 + 2 * HID_,
      hg1, NG_, HID_, HID_);
  gemm_wmma_kernel<<<gD0, 128, 0, stream>>>(
      aggGD, degGD, Wn1 + 1 * HID_ * HID_,
      hd0, nullptr, Ws1 + 1 * HID_ * HID_,
      nullptr, nullptr, nullptr,
      nullptr, nullptr, nullptr,
      b1 + 1 * HID_, nullptr,
      hd1, ND_, HID_, HID_);

  bn(hg1, NG_, bng + 2 * HID_, bnb + 2 * HID_);   // layer1, gene
  bn(hd1, ND_, bng + 3 * HID_, bnb + 3 * HID_);   // layer1, disease

  // ---- final projections (K=128, OUT=64) straight into d_out
  dim3 gGp((NG_ / 16 + 3) / 4, INF_ / 16);
  gemm_wmma_kernel<<<gGp, 128, 0, stream>>>(
      hg1, nullptr, Wp + 0 * HID_ * INF_,
      nullptr, nullptr, nullptr,
      nullptr, nullptr, nullptr,
      nullptr, nullptr, nullptr,
      bp + 0 * INF_, nullptr,
      outG, NG_, HID_, INF_);
  dim3 gDp((ND_ / 16 + 3) / 4, INF_ / 16);
  gemm_wmma_kernel<<<gDp, 128, 0, stream>>>(
      hd1, nullptr, Wp + 1 * HID_ * INF_,
      nullptr, nullptr, nullptr,
      nullptr, nullptr, nullptr,
      nullptr, nullptr, nullptr,
      bp + 1 * INF_, nullptr,
      outD, ND_, HID_, INF_);
}